// HardnessAwareKernelizedSupCon_78889959293669
// MI455X (gfx1250) — compile-verified
//
#include <hip/hip_runtime.h>
#include <math.h>

// N = 4096 rows (2048 batch x 2 views, view-major), D = 128
#define NROWS 4096
#define DIM   128
#define NB    2048
#define INV_T 14.285714285714286f   // 1 / 0.07
#define EPSF  1e-8f

typedef __attribute__((ext_vector_type(2))) float v2f;
typedef __attribute__((ext_vector_type(8))) float v8f;

// ds_swizzle_b32 group-of-32 XOR patterns: offset = and(0x1f) | xor_mask<<10
#define SWZ_X1  0x041F
#define SWZ_X2  0x081F
#define SWZ_X4  0x101F
#define SWZ_X8  0x201F
#define SWZ_X16 0x401F
#define SWZF(v, imm) __int_as_float(__builtin_amdgcn_ds_swizzle(__float_as_int(v), (imm)))

__device__ __forceinline__ float wave_reduce_add(float v) {
  v += SWZF(v, SWZ_X1);  v += SWZF(v, SWZ_X2);
  v += SWZF(v, SWZ_X4);  v += SWZF(v, SWZ_X8);
  v += SWZF(v, SWZ_X16);
  return v;
}
__device__ __forceinline__ float wave_reduce_min(float v) {
  v = fminf(v, SWZF(v, SWZ_X1));  v = fminf(v, SWZF(v, SWZ_X2));
  v = fminf(v, SWZF(v, SWZ_X4));  v = fminf(v, SWZF(v, SWZ_X8));
  v = fminf(v, SWZF(v, SWZ_X16));
  return v;
}
__device__ __forceinline__ float wave_reduce_max(float v) {
  v = fmaxf(v, SWZF(v, SWZ_X1));  v = fmaxf(v, SWZF(v, SWZ_X2));
  v = fmaxf(v, SWZF(v, SWZ_X4));  v = fmaxf(v, SWZF(v, SWZ_X8));
  v = fmaxf(v, SWZF(v, SWZ_X16));
  return v;
}

// A 16x4 f32 frag and B 4x16 f32 frag share the same indexing for S = Z*Z^T:
// lane -> M (or N) = lane&15 ; K pair = k0 + 2*(lane>>4) + {0,1} -> one b64 load
__device__ __forceinline__ v2f load_frag(const float* __restrict__ zn, int base, int k0, int lane) {
  int m = lane & 15, h = lane >> 4;
  return *(const v2f*)(zn + (base + m) * DIM + k0 + 2 * h);
}

// ---------------- Kernel 1: L2-normalize rows, build view-stacked z and y ---
__global__ void hk_normalize(const float* __restrict__ feats,
                             const float* __restrict__ labels,
                             float* __restrict__ zn, float* __restrict__ yv) {
  int n = blockIdx.x;          // 0..4095, view-major: n = v*2048 + b
  int t = threadIdx.x;         // 0..127
  int v = n >> 11, b = n & 2047;
  float x = feats[(b * 2 + v) * DIM + t];
  float ss = wave_reduce_add(x * x);
  __shared__ float sw[4];
  int w = t >> 5, l = t & 31;
  if (l == 0) sw[w] = ss;
  __syncthreads();
  float tot = sw[0] + sw[1] + sw[2] + sw[3];
  float inv = 1.0f / fmaxf(sqrtf(tot), 1e-12f);
  zn[n * DIM + t] = x * inv;
  if (t == 0) yv[n] = labels[b];
}

// ---------------- Kernel 2: GEMM pass 1 — global min(S) per block tile -------
__global__ void hk_gemm_min(const float* __restrict__ zn, float* __restrict__ bmin) {
  int lane = threadIdx.x & 31, wid = threadIdx.x >> 5;
  int rowBase = blockIdx.y * 16;
  int colBase = blockIdx.x * 128 + wid * 16;

  v8f acc = {0.f, 0.f, 0.f, 0.f, 0.f, 0.f, 0.f, 0.f};
#pragma unroll 4
  for (int k0 = 0; k0 < DIM; k0 += 4) {
    v2f a = load_frag(zn, rowBase, k0, lane);
    v2f b = load_frag(zn, colBase, k0, lane);
    acc = __builtin_amdgcn_wmma_f32_16x16x4_f32(false, a, false, b,
                                                (short)0, acc, false, false);
  }
  float m = acc[0];
#pragma unroll
  for (int r = 1; r < 8; ++r) m = fminf(m, acc[r]);
  m = wave_reduce_min(m);
  __shared__ float sm[8];
  if (lane == 0) sm[wid] = m;
  __syncthreads();
  if (threadIdx.x == 0) {
    float mm = sm[0];
#pragma unroll
    for (int i = 1; i < 8; ++i) mm = fminf(mm, sm[i]);
    bmin[blockIdx.y * gridDim.x + blockIdx.x] = mm;
  }
}

// ---------------- Kernel 3: reduce scalars: 1/(dy_max+eps), 1/(dz_max+eps) --
__global__ void hk_scalars(const float* __restrict__ bmin,
                           const float* __restrict__ labels,
                           float* __restrict__ scal) {
  int t = threadIdx.x, lane = t & 31, w = t >> 5;
  float mn = 3.0e38f;
  for (int i = t; i < 8192; i += 256) mn = fminf(mn, bmin[i]);
  float lmin = 3.0e38f, lmax = -3.0e38f;
  for (int i = t; i < NB; i += 256) {
    float y = labels[i];
    lmin = fminf(lmin, y); lmax = fmaxf(lmax, y);
  }
  mn = wave_reduce_min(mn);
  lmin = wave_reduce_min(lmin);
  lmax = wave_reduce_max(lmax);
  __shared__ float s0[8], s1[8], s2[8];
  if (lane == 0) { s0[w] = mn; s1[w] = lmin; s2[w] = lmax; }
  __syncthreads();
  if (t == 0) {
    float a = s0[0], b = s1[0], c = s2[0];
#pragma unroll
    for (int i = 1; i < 8; ++i) {
      a = fminf(a, s0[i]); b = fminf(b, s1[i]); c = fmaxf(c, s2[i]);
    }
    scal[0] = 1.0f / ((c - b) + EPSF);        // 1 / (max|dy| + eps)
    scal[1] = 1.0f / ((1.0f - a) + EPSF);     // 1 / (max d_z + eps)
  }
}

// ---------- Kernel 4: GEMM pass 2 — i on lanes (N), k on VGPR slots (M) -----
// All per-element contributions are symmetric in (i,k), so summing over k for
// fixed i == summing over the M direction == in-lane adds over the 8 acc VGPRs.
// Each block covers 128 i (8 waves x 16 lanes) x 128 k (8 chained 16-k tiles).
__global__ void hk_gemm_loss(const float* __restrict__ zn,
                             const float* __restrict__ yv,
                             const float* __restrict__ scal,
                             float* __restrict__ gA, float* __restrict__ gB,
                             float* __restrict__ gC) {
  int lane = threadIdx.x & 31, wid = threadIdx.x >> 5;
  int half = lane >> 4;
  int iBase = blockIdx.x * 128 + wid * 16;    // B/N side: loss index i
  int kSuper = blockIdx.y;                    // 128 k's per block

  float inv_dy = scal[0], inv_dz = scal[1];
  int iGlob = iBase + (lane & 15);
  float yi = yv[iGlob];

  float aAcc = 0.f, bAcc = 0.f, cAcc = 0.f;

#pragma unroll 1
  for (int kt = 0; kt < 8; ++kt) {
    int kBase = kSuper * 128 + kt * 16;       // A/M side: neighbor index k

    v8f acc = {0.f, 0.f, 0.f, 0.f, 0.f, 0.f, 0.f, 0.f};
#pragma unroll 4
    for (int k0 = 0; k0 < DIM; k0 += 4) {
      v2f a = load_frag(zn, kBase, k0, lane);
      v2f b = load_frag(zn, iBase, k0, lane);
      acc = __builtin_amdgcn_wmma_f32_16x16x4_f32(false, a, false, b,
                                                  (short)0, acc, false, false);
    }

    // C-layout: vgpr r, half h -> M = r + 8h (the k offset); N = lane&15 (i)
#pragma unroll
    for (int r = 0; r < 8; ++r) {
      int kGlob = kBase + r + 8 * half;
      float yk = yv[kGlob];
      float s  = acc[r];

      float dy = yi - yk;
      float wk = __expf(-2.0f * dy * dy);               // exp(-dy^2/(2*0.5^2))
      wk = fminf(fmaxf(wk, 0.0f), 1.0f);
      float d_y = fabsf(dy) * inv_dy;
      float d_z = (1.0f - s) * inv_dz;
      float u  = fminf(fmaxf(d_z - d_y, -1.0f), 1.0f);
      float m  = 1.0f + fmaxf(u, 0.0f);
      float nn = 1.0f + fmaxf(-u, 0.0f);
      float wh = wk * m + (1.0f - wk) * nn;
      float sim = s * INV_T;
      float mask = (iGlob != kGlob) ? 1.0f : 0.0f;

      aAcc += mask * __expf(sim);      // -> denom A
      bAcc += mask * wh * wh * sim;    // -> B
      cAcc += mask * wh;               // -> C
    }
  }

  // halves hold disjoint k subsets for the same i -> single xor-16 fold each
  aAcc += SWZF(aAcc, SWZ_X16);
  bAcc += SWZF(bAcc, SWZ_X16);
  cAcc += SWZF(cAcc, SWZ_X16);

  if (lane < 16) {
    int gi = iGlob * 32 + kSuper;
    gA[gi] = aAcc; gB[gi] = bAcc; gC[gi] = cAcc;
  }
}

// ---------------- Kernel 5: per-row loss term + partial sums ----------------
__global__ void hk_rowloss(const float* __restrict__ gA, const float* __restrict__ gB,
                           const float* __restrict__ gC, float* __restrict__ rpart) {
  int row = blockIdx.x * 256 + threadIdx.x;
  float A = 0.f, B = 0.f, C = 0.f;
#pragma unroll 4
  for (int cb = 0; cb < 32; ++cb) {
    A += gA[row * 32 + cb];
    B += gB[row * 32 + cb];
    C += gC[row * 32 + cb];
  }
  // sum_k pos_mask*log_prob / (sum_k pos_mask + eps)
  float val = (B - __logf(A + EPSF) * C) / (C + EPSF);
  val = wave_reduce_add(val);
  __shared__ float sv[8];
  int lane = threadIdx.x & 31, w = threadIdx.x >> 5;
  if (lane == 0) sv[w] = val;
  __syncthreads();
  if (threadIdx.x == 0) {
    float t = 0.f;
#pragma unroll
    for (int i = 0; i < 8; ++i) t += sv[i];
    rpart[blockIdx.x] = t;
  }
}

__global__ void hk_final(const float* __restrict__ rpart, float* __restrict__ out) {
  if (threadIdx.x == 0) {
    float t = 0.f;
#pragma unroll
    for (int i = 0; i < 16; ++i) t += rpart[i];
    out[0] = -t / (float)NROWS;   // T/base_T == 1
  }
}

extern "C" void kernel_launch(void* const* d_in, const int* in_sizes, int n_in,
                              void* d_out, int out_size, void* d_ws, size_t ws_size,
                              hipStream_t stream) {
  (void)in_sizes; (void)n_in; (void)out_size; (void)ws_size;
  const float* feats  = (const float*)d_in[0];   // [2048, 2, 128] f32
  const float* labels = (const float*)d_in[1];   // [2048, 1] f32
  float* out = (float*)d_out;

  float* ws    = (float*)d_ws;
  float* zn    = ws;                 // 4096*128        = 524288 floats
  float* yv    = ws + 524288;        // 4096
  float* bmin  = ws + 528384;        // 8192 block minima
  float* scal  = ws + 536576;        // 4 scalars (padded)
  float* gA    = ws + 536580;        // 4096*32
  float* gB    = gA + 131072;
  float* gC    = gB + 131072;
  float* rpart = gC + 131072;        // 16

  hk_normalize<<<NROWS, 128, 0, stream>>>(feats, labels, zn, yv);
  hk_gemm_min<<<dim3(32, 256), 256, 0, stream>>>(zn, bmin);
  hk_scalars<<<1, 256, 0, stream>>>(bmin, labels, scal);
  hk_gemm_loss<<<dim3(32, 32), 256, 0, stream>>>(zn, yv, scal, gA, gB, gC);
  hk_rowloss<<<16, 256, 0, stream>>>(gA, gB, gC, rpart);
  hk_final<<<1, 64, 0, stream>>>(rpart, out);
}